// IntraPredictor_12481174962473
// MI455X (gfx1250) — compile-verified
//
#include <hip/hip_runtime.h>

// u_dot_v gather-dot for MI455X (gfx1250, wave32).
// Pipeline per wave: GLOBAL_LOAD_ASYNC_TO_LDS_B128 streams 16 hu + 16 hv rows
// (512B coalesced per row) into a double-buffered LDS slab while the previous
// tile runs a 32x V_WMMA_F32_16X16X4_F32 chain (K=128, exact f32 accumulate).
// Diagonal of the 16x16 all-pairs result = the 16 per-edge scores.

typedef __attribute__((ext_vector_type(2))) float v2f;
typedef __attribute__((ext_vector_type(8))) float v8f;

#define D_FEAT          128
#define ROW_STRIDE      132                  // 128 + 4 pad floats: column reads hit banks 0,4,...,60
#define MAT_FLOATS      (16 * ROW_STRIDE)    // one 16-row tile (hu or hv)
#define BUF_FLOATS      (2 * MAT_FLOATS)     // hu + hv
#define TILES_PER_WAVE  4                    // 64 edges per wave; 37500 tiles / 4 = 9375 blocks, no tail

__global__ __launch_bounds__(32)
void u_dot_v_wmma_async(const float* __restrict__ h,
                        const int*   __restrict__ src,
                        const int*   __restrict__ dst,
                        float*       __restrict__ out,
                        int n_edges, int n_nodes)
{
    __shared__ float smem[2 * BUF_FLOATS];   // double buffer, 33792 B, one wave per block

    const int lane = threadIdx.x;            // 0..31
    const unsigned lds_base = (unsigned)(size_t)&smem[0];   // addr[31:0] == LDS byte offset
    const unsigned lane16   = (unsigned)lane * 16u;
    const unsigned long long hbase = (unsigned long long)(const void*)h;

    const int tile0 = blockIdx.x * TILES_PER_WAVE;

    // Issue one tile's 32 async row transfers (16 hu rows + 16 hv rows, 512B each).
    auto issue_tile = [&](int t, int buf) {
        const int e0 = (tile0 + t) * 16;
        int e = e0 + (lane & 15);
        if (e > n_edges - 1) e = n_edges - 1;            // clamp: keeps EXEC full, stores guarded later
        int node = (lane < 16) ? src[e] : dst[e];
        if (node < 0) node = 0;
        if (node > n_nodes - 1) node = n_nodes - 1;

        const unsigned ldsA = lds_base + (unsigned)buf * (BUF_FLOATS * 4u) + lane16;
        const unsigned ldsB = ldsA + MAT_FLOATS * 4u;
        #pragma unroll
        for (int r = 0; r < 16; ++r) {
            int su = __shfl(node, r, 32);                // src node of edge r
            int sv = __shfl(node, r + 16, 32);           // dst node of edge r
            unsigned long long ga = hbase + (unsigned long long)(unsigned)su * (D_FEAT * 4ull) + lane16;
            unsigned long long gb = hbase + (unsigned long long)(unsigned)sv * (D_FEAT * 4ull) + lane16;
            unsigned la = ldsA + (unsigned)r * (ROW_STRIDE * 4u);
            unsigned lb = ldsB + (unsigned)r * (ROW_STRIDE * 4u);
            asm volatile("global_load_async_to_lds_b128 %0, %1, off"
                         :: "v"(la), "v"(ga) : "memory");
            asm volatile("global_load_async_to_lds_b128 %0, %1, off"
                         :: "v"(lb), "v"(gb) : "memory");
        }
    };

    issue_tile(0, 0);                                    // prologue: fill buffer 0

    const int m    = lane & 15;
    const int ksel = (lane < 16) ? 0 : 2;

    for (int t = 0; t < TILES_PER_WAVE; ++t) {
        const int buf = t & 1;

        if (t + 1 < TILES_PER_WAVE) {
            issue_tile(t + 1, buf ^ 1);                  // overlap next tile's gather
            // async loads complete in order: <=32 outstanding => tile t's 32 are done
            asm volatile("s_wait_asynccnt 0x20" ::: "memory");
        } else {
            asm volatile("s_wait_asynccnt 0x0" ::: "memory");
        }

        // ---- WMMA chain: D(16x16) += A(16x4)*B(4x16), 32 steps cover K=128
        // A frag (hu): lane L -> row m=L%16, K=(L<16?0:2)+{0,1}; B frag (hv): same addressing on hv tile
        const float* LA = smem + buf * BUF_FLOATS;
        const float* LB = LA + MAT_FLOATS;
        const float* pA = LA + m * ROW_STRIDE + ksel;
        const float* pB = LB + m * ROW_STRIDE + ksel;

        v8f acc = {};
        #pragma unroll
        for (int k = 0; k < 32; ++k) {
            v2f a = *(const v2f*)(pA + k * 4);           // 8B-aligned ds_load_b64
            v2f b = *(const v2f*)(pB + k * 4);
            acc = __builtin_amdgcn_wmma_f32_16x16x4_f32(
                      /*neg_a=*/false, a, /*neg_b=*/false, b,
                      /*c_mod=*/(short)0, acc, /*reuse_a=*/false, /*reuse_b=*/false);
        }

        // ---- diagonal extraction: lanes 0-7 hold M=0..7 in acc[lane], lanes 24-31 hold M=8..15 in acc[lane-24]
        const bool writer = (lane < 8) || (lane >= 24);
        int di = ((lane < 8) ? lane : (lane - 24)) & 7;
        float s = acc[0];
        s = (di == 1) ? acc[1] : s;
        s = (di == 2) ? acc[2] : s;
        s = (di == 3) ? acc[3] : s;
        s = (di == 4) ? acc[4] : s;
        s = (di == 5) ? acc[5] : s;
        s = (di == 6) ? acc[6] : s;
        s = (di == 7) ? acc[7] : s;

        const int e0   = (tile0 + t) * 16;
        const int eout = e0 + ((lane < 8) ? lane : (lane - 16));
        if (writer && eout < n_edges) out[eout] = s;
    }
}

extern "C" void kernel_launch(void* const* d_in, const int* in_sizes, int n_in,
                              void* d_out, int out_size, void* d_ws, size_t ws_size,
                              hipStream_t stream)
{
    const float* h   = (const float*)d_in[0];
    const int*   src = (const int*)d_in[1];
    const int*   dst = (const int*)d_in[2];
    float*       out = (float*)d_out;

    const int n_nodes = in_sizes[0] / D_FEAT;
    const int n_edges = in_sizes[1];

    const int edges_per_block = TILES_PER_WAVE * 16;     // 64
    const int grid = (n_edges + edges_per_block - 1) / edges_per_block;

    u_dot_v_wmma_async<<<grid, 32, 0, stream>>>(h, src, dst, out, n_edges, n_nodes);
}